// ContrastiveLoss_21388937134644
// MI455X (gfx1250) — compile-verified
//
#include <hip/hip_runtime.h>
#include <math.h>

typedef __attribute__((ext_vector_type(2))) float v2f;
typedef __attribute__((ext_vector_type(8))) float v8f;

#define DIMD 64
#define BM 128
#define BN 128
#define LDSS 68   // padded LDS row stride (floats): banks = 4*idx+col -> conflict-free

// ---------------- Kernel 1: row squared norms + reset max accumulator ----------------
__global__ void mpd_norms_kernel(const float* __restrict__ L, const float* __restrict__ R,
                                 float* __restrict__ lsq, float* __restrict__ rsq,
                                 unsigned int* __restrict__ maxbits, int NL, int NR) {
  int t = blockIdx.x * blockDim.x + threadIdx.x;
  if (t == 0) *maxbits = 0u;   // 0u == 0.0f bits; all candidates clamped >= 0
  if (t >= NL + NR) return;
  const float* src; float* dst; int row;
  if (t < NL) { src = L; dst = lsq; row = t; }
  else        { src = R; dst = rsq; row = t - NL; }
  const float4* p = (const float4*)(src + (size_t)row * DIMD);
  float s = 0.f;
#pragma unroll
  for (int i = 0; i < DIMD / 4; ++i) {
    float4 v = p[i];
    s += v.x * v.x + v.y * v.y + v.z * v.z + v.w * v.w;
  }
  dst[row] = s;
}

// ---------------- Kernel 2: WMMA fp32 GEMM (L @ R^T) fused with max(l2+r2-2c) ----------------
__global__ __launch_bounds__(256, 2) void mpd_gemm_max_kernel(
    const float* __restrict__ L, const float* __restrict__ R,
    const float* __restrict__ lsq, const float* __restrict__ rsq,
    unsigned int* __restrict__ maxbits) {
  __shared__ float sL[BM * LDSS];
  __shared__ float sR[BN * LDSS];

  const int tid    = threadIdx.x;
  const int blockM = blockIdx.x * BM;
  const int blockN = blockIdx.y * BN;

  // Stage both 128x64 fp32 tiles into LDS (2048 float4 each; 8 per thread).
#pragma unroll
  for (int i = 0; i < 8; ++i) {
    int idx = i * 256 + tid;   // 0..2047
    int row = idx >> 4;
    int c4  = idx & 15;
    float4 vl = ((const float4*)(L + (size_t)(blockM + row) * DIMD))[c4];
    float4 vr = ((const float4*)(R + (size_t)(blockN + row) * DIMD))[c4];
    *(float4*)&sL[row * LDSS + c4 * 4] = vl;
    *(float4*)&sR[row * LDSS + c4 * 4] = vr;
  }
  __syncthreads();

  const int lane = tid & 31;
  const int wid  = tid >> 5;    // 8 waves
  const int wm   = wid & 1;     // 2 waves along M
  const int wn   = wid >> 1;    // 4 waves along N
  const int hf   = lane >> 4;   // half of the wave (lanes 0-15 / 16-31)
  const int li   = lane & 15;

  const int wMBase = wm * 64;   // wave computes 64x32
  const int wNBase = wn * 32;

  v8f acc[4][2];
#pragma unroll
  for (int mt = 0; mt < 4; ++mt)
#pragma unroll
    for (int nt = 0; nt < 2; ++nt)
      acc[mt][nt] = (v8f){0.f, 0.f, 0.f, 0.f, 0.f, 0.f, 0.f, 0.f};

  // K = 64 in 16 steps of K=4. A layout: lane(li,hf) holds K = 2*hf + {0,1}.
  // B[k][n] = R[n][k] -> same 8-byte row-major fragment load from sR.
#pragma unroll
  for (int s = 0; s < 16; ++s) {
    const int col = 4 * s + 2 * hf;
    v2f a[4], b[2];
#pragma unroll
    for (int mt = 0; mt < 4; ++mt)
      a[mt] = *(const v2f*)&sL[(wMBase + mt * 16 + li) * LDSS + col];
#pragma unroll
    for (int nt = 0; nt < 2; ++nt)
      b[nt] = *(const v2f*)&sR[(wNBase + nt * 16 + li) * LDSS + col];
#pragma unroll
    for (int mt = 0; mt < 4; ++mt)
#pragma unroll
      for (int nt = 0; nt < 2; ++nt)
        acc[mt][nt] = __builtin_amdgcn_wmma_f32_16x16x4_f32(
            false, a[mt], false, b[nt], (short)0, acc[mt][nt], false, false);
  }

  // Epilogue: C/D layout -> VGPR i is M = 8*hf + i, lane li is N = li.
  float m = 0.f;
#pragma unroll
  for (int mt = 0; mt < 4; ++mt) {
    const int M0 = blockM + wMBase + mt * 16 + 8 * hf;   // multiple of 8 -> aligned float4
    float4 l0 = *(const float4*)&lsq[M0];
    float4 l1 = *(const float4*)&lsq[M0 + 4];
    float lv[8] = {l0.x, l0.y, l0.z, l0.w, l1.x, l1.y, l1.z, l1.w};
#pragma unroll
    for (int nt = 0; nt < 2; ++nt) {
      const float rv = rsq[blockN + wNBase + nt * 16 + li];
#pragma unroll
      for (int i = 0; i < 8; ++i) {
        float v = lv[i] + rv - 2.f * acc[mt][nt][i];
        m = fmaxf(m, v);   // also clamps tiny negatives via init 0
      }
    }
  }
  // wave32 max-reduce, then one atomic per wave (uint order == float order for >= 0)
#pragma unroll
  for (int off = 16; off >= 1; off >>= 1)
    m = fmaxf(m, __shfl_xor(m, off, 32));
  if (lane == 0)
    atomicMax(maxbits, __float_as_uint(m));
}

// ---------------- Kernel 3: finalize ----------------
__global__ void mpd_finish_kernel(const unsigned int* __restrict__ maxbits,
                                  float* __restrict__ out) {
  out[0] = sqrtf(__uint_as_float(*maxbits));
}

extern "C" void kernel_launch(void* const* d_in, const int* in_sizes, int n_in,
                              void* d_out, int out_size, void* d_ws, size_t ws_size,
                              hipStream_t stream) {
  const float* L = (const float*)d_in[0];
  const float* R = (const float*)d_in[1];
  const int NL = in_sizes[0] / DIMD;
  const int NR = in_sizes[1] / DIMD;

  float* w = (float*)d_ws;
  float* lsq = w;
  float* rsq = w + NL;
  unsigned int* maxbits = (unsigned int*)(w + NL + NR);

  const int total = NL + NR;
  mpd_norms_kernel<<<(total + 255) / 256, 256, 0, stream>>>(L, R, lsq, rsq, maxbits, NL, NR);

  dim3 grid(NL / BM, NR / BN);
  mpd_gemm_max_kernel<<<grid, 256, 0, stream>>>(L, R, lsq, rsq, maxbits);

  mpd_finish_kernel<<<1, 1, 0, stream>>>(maxbits, (float*)d_out);
}